// LiftedStructureLoss_652835029339
// MI455X (gfx1250) — compile-verified
//
#include <hip/hip_runtime.h>
#include <hip/hip_bf16.h>

typedef __attribute__((ext_vector_type(2))) float v2f;
typedef __attribute__((ext_vector_type(8))) float v8f;

#define BN       4096
#define DK       128
#define KCHUNKS  32                 // WMMA k-steps (K=4 each) per 16x16 tile
#define KHALF    16
#define NTILES   (BN / 16)          // 256
#define JGROUPS  16
#define TPG      (NTILES / JGROUPS) // 16 j-tiles per block

#define WMMA_F32(acc, A, Bv) \
    __builtin_amdgcn_wmma_f32_16x16x4_f32(false, (A), false, (Bv), (short)0, (acc), false, false)

// ---------------------------------------------------------------------------
// Kernel 0: per-row squared magnitude; zero nes + accumulators (ws is poisoned)
// ---------------------------------------------------------------------------
__global__ void lsl_init_kernel(const float* __restrict__ score,
                                float* __restrict__ mag,
                                float* __restrict__ nes,
                                float* __restrict__ acc) {
    const int row  = blockIdx.x;
    const int lane = threadIdx.x;
    const float* r = score + (size_t)row * DK;
    float s = 0.f;
    #pragma unroll
    for (int k = lane; k < DK; k += 32) {
        float v = r[k];
        s += v * v;
    }
    #pragma unroll
    for (int m = 16; m >= 1; m >>= 1) s += __shfl_xor(s, m, 32);
    if (lane == 0) {
        mag[row] = s;
        nes[row] = 0.f;
        if (row == 0) { acc[0] = 0.f; acc[1] = 0.f; }
    }
}

// ---------------------------------------------------------------------------
// Kernel 1: nes[i] = sum_j [tgt_i != tgt_j] * exp(alpha - dist(i,j))
// One wave per (16-row strip, j-group). A strip staged in LDS (WMMA A-layout,
// bank-conflict-free), B tile register-resident + half-tile software pipeline.
// ---------------------------------------------------------------------------
__global__ __launch_bounds__(32)
void lsl_nes_kernel(const float* __restrict__ score,
                    const int*   __restrict__ tgt,
                    const float* __restrict__ mag,
                    const float* __restrict__ alpha,
                    float* __restrict__ nes) {
    const int lane  = threadIdx.x;   // 0..31, full wave32
    const int m16   = lane & 15;
    const int hi    = lane >> 4;     // 0 or 1
    const int i0    = blockIdx.x * 16;
    const int jg    = blockIdx.y;
    const int jbase = jg * TPG;      // first j-tile index for this block

    // Stage A strip in LDS, already in the 16x4 f32 WMMA A layout:
    // chunk c, lane l -> ldsA[c*32 + l]  (8B stride across lanes: conflict-free)
    __shared__ v2f ldsA[KCHUNKS * 32];
    {
        const float* arow = score + (size_t)(i0 + m16) * DK + hi * 2;
        #pragma unroll
        for (int c = 0; c < KCHUNKS; ++c)
            ldsA[c * 32 + lane] = *(const v2f*)(arow + c * 4);
    }
    __syncthreads();   // single-wave WG: barrier is a NOP; LDS same-wave ops are in-order

    const float alphav = alpha[0];

    float magi[8];
    int   ti[8];
    #pragma unroll
    for (int r = 0; r < 8; ++r) {
        const int i = i0 + r + 8 * hi;   // C/D layout: VGPR r, lane half -> M
        magi[r] = mag[i];
        ti[r]   = tgt[i];
    }

    float nacc[8];
    #pragma unroll
    for (int r = 0; r < 8; ++r) nacc[r] = 0.f;

    // Preload B tile for jt=0 (register-resident, batched loads).
    v2f b[KCHUNKS];
    {
        const float* brow = score + (size_t)(jbase * 16 + m16) * DK + hi * 2;
        #pragma unroll
        for (int c = 0; c < KCHUNKS; ++c)
            b[c] = *(const v2f*)(brow + c * 4);
    }

    #pragma unroll 1
    for (int jt = 0; jt < TPG; ++jt) {
        const int    j         = (jbase + jt) * 16 + m16;
        const float* brow_next = score + (size_t)(j + 16) * DK + hi * 2;
        const bool   have_next = (jt + 1 < TPG);

        const float magj = mag[j];
        const int   tj   = tgt[j];

        v8f accv = {};
        // First K-half WMMAs (A chunks streamed from LDS) ...
        #pragma unroll
        for (int c = 0; c < KHALF; ++c)
            accv = WMMA_F32(accv, ldsA[c * 32 + lane], b[c]);
        // ... while refilling the consumed B half with the next tile's data.
        if (have_next) {
            #pragma unroll
            for (int c = 0; c < KHALF; ++c)
                b[c] = *(const v2f*)(brow_next + c * 4);
        }
        // Second K-half WMMAs ...
        #pragma unroll
        for (int c = KHALF; c < KCHUNKS; ++c)
            accv = WMMA_F32(accv, ldsA[c * 32 + lane], b[c]);
        if (have_next) {
            #pragma unroll
            for (int c = KHALF; c < KCHUNKS; ++c)
                b[c] = *(const v2f*)(brow_next + c * 4);
        }

        // Epilogue: dist + masked exp, overlaps next tile's in-flight loads.
        #pragma unroll
        for (int r = 0; r < 8; ++r) {
            float d2   = magi[r] + magj - 2.0f * accv[r];
            float dist = sqrtf(fmaxf(d2, 0.0f));
            nacc[r] += (ti[r] != tj) ? __expf(alphav - dist) : 0.0f;
        }
    }

    // Reduce over the 16 lanes of each half (columns), one atomic per row.
    #pragma unroll
    for (int r = 0; r < 8; ++r) {
        float v = nacc[r];
        v += __shfl_xor(v, 1, 32);
        v += __shfl_xor(v, 2, 32);
        v += __shfl_xor(v, 4, 32);
        v += __shfl_xor(v, 8, 32);
        if (m16 == 0)
            atomicAdd(&nes[i0 + r + 8 * hi], v);
    }
}

// ---------------------------------------------------------------------------
// Kernel 2: sum over positive pairs (i<j, equal label) of
//           relu(log(nes_i + nes_j) + dist)^2 , plus pair count.
// Uniform triangle skip via computed start tile (EXEC stays all-1s).
// ---------------------------------------------------------------------------
__global__ __launch_bounds__(32)
void lsl_loss_kernel(const float* __restrict__ score,
                     const int*   __restrict__ tgt,
                     const float* __restrict__ mag,
                     const float* __restrict__ nes,
                     float* __restrict__ acc) {
    const int lane  = threadIdx.x;
    const int m16   = lane & 15;
    const int hi    = lane >> 4;
    const int bi    = blockIdx.x;     // i-tile index
    const int i0    = bi * 16;
    const int jg    = blockIdx.y;
    const int jbase = jg * TPG;

    // keep j-tiles with tile index >= bi (others are strictly below diagonal)
    int jt0 = bi - jbase;
    if (jt0 < 0) jt0 = 0;
    if (jt0 >= TPG) return;           // whole group below diagonal (uniform)

    __shared__ v2f ldsA[KCHUNKS * 32];
    {
        const float* arow = score + (size_t)(i0 + m16) * DK + hi * 2;
        #pragma unroll
        for (int c = 0; c < KCHUNKS; ++c)
            ldsA[c * 32 + lane] = *(const v2f*)(arow + c * 4);
    }
    __syncthreads();

    float magi[8], nesi[8];
    int   ti[8];
    #pragma unroll
    for (int r = 0; r < 8; ++r) {
        const int i = i0 + r + 8 * hi;
        magi[r] = mag[i];
        nesi[r] = nes[i];
        ti[r]   = tgt[i];
    }

    float lsum = 0.f;
    float lcnt = 0.f;

    v2f b[KCHUNKS];
    {
        const float* brow = score + (size_t)((jbase + jt0) * 16 + m16) * DK + hi * 2;
        #pragma unroll
        for (int c = 0; c < KCHUNKS; ++c)
            b[c] = *(const v2f*)(brow + c * 4);
    }

    #pragma unroll 1
    for (int jt = jt0; jt < TPG; ++jt) {
        const int    j         = (jbase + jt) * 16 + m16;
        const float* brow_next = score + (size_t)(j + 16) * DK + hi * 2;
        const bool   have_next = (jt + 1 < TPG);

        const float magj = mag[j];
        const float nesj = nes[j];
        const int   tj   = tgt[j];

        v8f accv = {};
        #pragma unroll
        for (int c = 0; c < KHALF; ++c)
            accv = WMMA_F32(accv, ldsA[c * 32 + lane], b[c]);
        if (have_next) {
            #pragma unroll
            for (int c = 0; c < KHALF; ++c)
                b[c] = *(const v2f*)(brow_next + c * 4);
        }
        #pragma unroll
        for (int c = KHALF; c < KCHUNKS; ++c)
            accv = WMMA_F32(accv, ldsA[c * 32 + lane], b[c]);
        if (have_next) {
            #pragma unroll
            for (int c = KHALF; c < KCHUNKS; ++c)
                b[c] = *(const v2f*)(brow_next + c * 4);
        }

        #pragma unroll
        for (int r = 0; r < 8; ++r) {
            const int  i   = i0 + r + 8 * hi;
            const bool pos = (ti[r] == tj) && (i < j);
            float d2   = magi[r] + magj - 2.0f * accv[r];
            float dist = sqrtf(fmaxf(d2, 0.0f));
            float t    = fmaxf(__logf(nesi[r] + nesj) + dist, 0.0f);
            lsum += pos ? t * t : 0.0f;
            lcnt += pos ? 1.0f : 0.0f;
        }
    }

    // full-wave reduction, two atomics per wave
    #pragma unroll
    for (int m = 16; m >= 1; m >>= 1) {
        lsum += __shfl_xor(lsum, m, 32);
        lcnt += __shfl_xor(lcnt, m, 32);
    }
    if (lane == 0) {
        atomicAdd(&acc[0], lsum);
        atomicAdd(&acc[1], lcnt);
    }
}

// ---------------------------------------------------------------------------
// Kernel 3: finalize
// ---------------------------------------------------------------------------
__global__ void lsl_final_kernel(const float* __restrict__ acc,
                                 float* __restrict__ out) {
    out[0] = acc[0] / (2.0f * acc[1]);
}

extern "C" void kernel_launch(void* const* d_in, const int* in_sizes, int n_in,
                              void* d_out, int out_size, void* d_ws, size_t ws_size,
                              hipStream_t stream) {
    const float* score = (const float*)d_in[0];
    const int*   tgt   = (const int*)d_in[1];
    const float* alpha = (const float*)d_in[2];
    float*       out   = (float*)d_out;

    // workspace layout: [0..4095] mag, [4096..8191] nes, [8192..8193] {sum, count}
    float* mag = (float*)d_ws;
    float* nes = mag + BN;
    float* acc = nes + BN;

    lsl_init_kernel<<<BN, 32, 0, stream>>>(score, mag, nes, acc);
    lsl_nes_kernel<<<dim3(NTILES, JGROUPS), 32, 0, stream>>>(score, tgt, mag, alpha, nes);
    lsl_loss_kernel<<<dim3(NTILES, JGROUPS), 32, 0, stream>>>(score, tgt, mag, nes, acc);
    lsl_final_kernel<<<1, 1, 0, stream>>>(acc, out);
}